// hierarchical_down_block_batch_17557826306190
// MI455X (gfx1250) — compile-verified
//
#include <hip/hip_runtime.h>
#include <hip/hip_bf16.h>

typedef float v2f __attribute__((ext_vector_type(2)));
typedef float v8f __attribute__((ext_vector_type(8)));

#define B_    4
#define IN_   32
#define OUT_  64
#define N_    163842
#define NS_   40962
#define K1_   224        // 7*IN
#define K2_   128        // 2*OUT
#define EPS_  1e-5f
#define SLOPE_ 0.2f
#define COLT_ ((NS_ + 15) / 16)   // 2561 column tiles of 16 points

// ---------------------------------------------------------------------------
// Kernel 1: 7-neighbor mean pool on the fine mesh.  pooled[b,c,s]
// ---------------------------------------------------------------------------
__global__ void pool_kernel(const float* __restrict__ x,
                            const int* __restrict__ pool_neigh,
                            float* __restrict__ pooled) {
    int i = blockIdx.x * blockDim.x + threadIdx.x;
    const int total = B_ * IN_ * NS_;
    if (i >= total) return;
    int s  = i % NS_;
    int bc = i / NS_;                      // b*IN + c
    const float* xb = x + (long)bc * N_;
    const int*   pn = pool_neigh + s * 7;
    float acc = 0.0f;
#pragma unroll
    for (int j = 0; j < 7; ++j) acc += xb[pn[j]];
    pooled[i] = acc * (1.0f / 7.0f);
}

// ---------------------------------------------------------------------------
// Kernel 2: zero the batchnorm accumulator (sum[64] | sumsq[64])
// ---------------------------------------------------------------------------
__global__ void zero_stats(float* __restrict__ stats) {
    if (threadIdx.x < 128) stats[threadIdx.x] = 0.0f;
}

// ---------------------------------------------------------------------------
// Kernel 3: one-ring conv as 64x224 GEMM via V_WMMA_F32_16X16X4_F32.
// Block = 128 threads = 4 waves; each wave owns a 16-row (channel) tile of the
// same 16-point column tile.  Gathered stencil staged in LDS.
// Also accumulates per-channel sum / sumsq for the batchnorm.
// ---------------------------------------------------------------------------
__global__ __launch_bounds__(128) void conv1_kernel(
    const float* __restrict__ pooled, const int* __restrict__ neigh,
    const float* __restrict__ W1, const float* __restrict__ b1,
    float* __restrict__ h, float* __restrict__ stats) {

    __shared__ float smem[K1_ * 16];               // [k][n], 14336 B

    const int tid  = threadIdx.x;
    const int wave = tid >> 5;
    const int lane = tid & 31;
    const int g    = lane >> 4;                    // half-wave: selects K pair / M+8
    const int m    = lane & 15;

    const int t       = blockIdx.x;
    const int b       = t / COLT_;
    const int colBase = (t % COLT_) * 16;

    // ---- stage gathered 224 x 16 operand tile into LDS ----
    for (int e = tid; e < K1_ * 16; e += 128) {
        int k = e >> 4;                            // 0..223  (k = j*32 + c)
        int n = e & 15;
        int s = colBase + n; if (s >= NS_) s = NS_ - 1;   // clamp (masked later)
        int j = k >> 5;                            // stencil slot 0..6
        int c = k & 31;                            // input channel
        int idx = neigh[s * 7 + j];                // coarse-mesh gather index
        smem[e] = pooled[((b * IN_) + c) * NS_ + idx];
    }
    __syncthreads();

    // ---- 56 WMMA K-steps of 4 ----
    const int rowBase = wave * 16;
    const float* wrow = W1 + (rowBase + m) * K1_ + 2 * g;  // A: K = v + 2*g
    v8f acc = {};
    for (int k0 = 0; k0 < K1_; k0 += 4) {
        v2f a, bmat;
        a.x    = wrow[k0];
        a.y    = wrow[k0 + 1];
        bmat.x = smem[(k0 + 2 * g) * 16 + m];      // B: K = v + 2*g, N = m
        bmat.y = smem[(k0 + 2 * g + 1) * 16 + m];
        acc = __builtin_amdgcn_wmma_f32_16x16x4_f32(
            false, a, false, bmat, (short)0, acc, false, false);
    }

    // ---- epilogue: bias, store h, accumulate BN stats ----
    const int s = colBase + m;
    const float valid = (s < NS_) ? 1.0f : 0.0f;
#pragma unroll
    for (int r = 0; r < 8; ++r) {
        const int ch = rowBase + r + 8 * g;        // C/D layout: M = r (+8 upper half)
        float v = acc[r] + b1[ch];
        if (s < NS_) h[((b * OUT_) + ch) * NS_ + s] = v;
        float sv = v * valid;
        float sq = v * v * valid;
        // reduce across the 16 lanes of this half-wave (masks < 16 stay in-half)
        sv += __shfl_xor(sv, 8); sv += __shfl_xor(sv, 4);
        sv += __shfl_xor(sv, 2); sv += __shfl_xor(sv, 1);
        sq += __shfl_xor(sq, 8); sq += __shfl_xor(sq, 4);
        sq += __shfl_xor(sq, 2); sq += __shfl_xor(sq, 1);
        if (m == 0) {
            atomicAdd(&stats[ch],      sv);
            atomicAdd(&stats[64 + ch], sq);
        }
    }
}

// ---------------------------------------------------------------------------
// Kernel 4: fold sums into per-channel BN scale/shift
// ---------------------------------------------------------------------------
__global__ void stats_kernel(const float* __restrict__ stats,
                             const float* __restrict__ gamma,
                             const float* __restrict__ beta,
                             float* __restrict__ scsh) {
    int o = threadIdx.x;
    if (o < OUT_) {
        const float inv = 1.0f / (float)(B_ * NS_);
        float mu  = stats[o] * inv;
        float var = stats[64 + o] * inv - mu * mu;      // biased var (ddof=0)
        float sc  = gamma[o] * rsqrtf(var + EPS_);
        scsh[o]        = sc;
        scsh[64 + o]   = beta[o] - mu * sc;
    }
}

// ---------------------------------------------------------------------------
// Kernel 5: BN + LeakyReLU + concat + 1x1 conv as 64x128 GEMM via WMMA.
// ---------------------------------------------------------------------------
__global__ __launch_bounds__(128) void conv2_kernel(
    const float* __restrict__ h, const float* __restrict__ x1,
    const float* __restrict__ Wc, const float* __restrict__ bc,
    const float* __restrict__ scsh, float* __restrict__ out) {

    __shared__ float smem[K2_ * 16];               // [k][n], 8192 B

    const int tid  = threadIdx.x;
    const int wave = tid >> 5;
    const int lane = tid & 31;
    const int g    = lane >> 4;
    const int m    = lane & 15;

    const int t       = blockIdx.x;
    const int b       = t / COLT_;
    const int colBase = (t % COLT_) * 16;

    // ---- stage concatenated [BN+LReLU(h) ; x1] 128 x 16 tile ----
    for (int e = tid; e < K2_ * 16; e += 128) {
        int k = e >> 4;                            // 0..127
        int n = e & 15;
        int s = colBase + n; if (s >= NS_) s = NS_ - 1;
        float v;
        if (k < OUT_) {
            float hv = h[((b * OUT_) + k) * NS_ + s];
            v = scsh[k] * hv + scsh[64 + k];
            v = (v >= 0.0f) ? v : SLOPE_ * v;      // LeakyReLU(0.2)
        } else {
            v = x1[((b * OUT_) + (k - OUT_)) * NS_ + s];
        }
        smem[e] = v;
    }
    __syncthreads();

    // ---- 32 WMMA K-steps of 4 ----
    const int rowBase = wave * 16;
    const float* wrow = Wc + (rowBase + m) * K2_ + 2 * g;
    v8f acc = {};
    for (int k0 = 0; k0 < K2_; k0 += 4) {
        v2f a, bmat;
        a.x    = wrow[k0];
        a.y    = wrow[k0 + 1];
        bmat.x = smem[(k0 + 2 * g) * 16 + m];
        bmat.y = smem[(k0 + 2 * g + 1) * 16 + m];
        acc = __builtin_amdgcn_wmma_f32_16x16x4_f32(
            false, a, false, bmat, (short)0, acc, false, false);
    }

    // ---- epilogue: bias + store ----
    const int s = colBase + m;
#pragma unroll
    for (int r = 0; r < 8; ++r) {
        const int ch = rowBase + r + 8 * g;
        if (s < NS_) out[((b * OUT_) + ch) * NS_ + s] = acc[r] + bc[ch];
    }
}

// ---------------------------------------------------------------------------
// Host-side launcher
// ---------------------------------------------------------------------------
extern "C" void kernel_launch(void* const* d_in, const int* in_sizes, int n_in,
                              void* d_out, int out_size, void* d_ws, size_t ws_size,
                              hipStream_t stream) {
    const float* x          = (const float*)d_in[0];
    const float* x1         = (const float*)d_in[1];
    const int*   pool_neigh = (const int*)  d_in[2];
    const int*   neigh      = (const int*)  d_in[3];
    const float* W1         = (const float*)d_in[4];
    const float* b1         = (const float*)d_in[5];
    const float* gamma      = (const float*)d_in[6];
    const float* beta       = (const float*)d_in[7];
    const float* Wc         = (const float*)d_in[8];
    const float* bc         = (const float*)d_in[9];
    float* out = (float*)d_out;

    // workspace layout (floats): pooled | h | stats(128) | scale+shift(128)
    float* pooled = (float*)d_ws;
    float* h      = pooled + (size_t)B_ * IN_ * NS_;
    float* stats  = h      + (size_t)B_ * OUT_ * NS_;
    float* scsh   = stats  + 128;

    // 1) pool
    {
        int total  = B_ * IN_ * NS_;
        int blocks = (total + 255) / 256;
        pool_kernel<<<blocks, 256, 0, stream>>>(x, pool_neigh, pooled);
    }
    // 2) zero BN accumulators
    zero_stats<<<1, 128, 0, stream>>>(stats);
    // 3) conv1 (WMMA) + BN stat accumulation
    conv1_kernel<<<B_ * COLT_, 128, 0, stream>>>(pooled, neigh, W1, b1, h, stats);
    // 4) BN scale/shift
    stats_kernel<<<1, 64, 0, stream>>>(stats, gamma, beta, scsh);
    // 5) BN+LReLU+concat+1x1 conv (WMMA)
    conv2_kernel<<<B_ * COLT_, 128, 0, stream>>>(h, x1, Wc, bc, scsh, out);
}